// PPFEmbeddingSin_44289702756526
// MI455X (gfx1250) — compile-verified
//
#include <hip/hip_runtime.h>
#include <stdint.h>

#ifndef __has_builtin
#define __has_builtin(x) 0
#endif

#if defined(__HIP_DEVICE_COMPILE__) && \
    __has_builtin(__builtin_amdgcn_global_load_async_to_lds_b32) && \
    __has_builtin(__builtin_amdgcn_s_wait_asynccnt)
#define USE_ASYNC_LDS 1
#else
#define USE_ASYNC_LDS 0
#endif

#define B_    4
#define N_    20000
#define M_    1024
#define D_    256
#define KNN   50
#define SLOTS 64
#define DROP  (SLOTS - KNN)      // 14
#define CHUNK 2048
#define WAVES 8
#define NPF   84                 // NUM_POS_FEATS

typedef float v2f __attribute__((ext_vector_type(2)));
typedef float v8f __attribute__((ext_vector_type(8)));

#if USE_ASYNC_LDS
typedef __attribute__((address_space(1))) int* gint_ptr;
typedef __attribute__((address_space(3))) int* lint_ptr;
#endif

// ---------------- wave32 reductions ----------------
__device__ __forceinline__ float wave_max32(float v) {
#pragma unroll
  for (int o = 16; o > 0; o >>= 1) v = fmaxf(v, __shfl_xor(v, o, 32));
  return v;
}

// argmax over wave of (m, packed-id); deterministic tie-break on packed id
__device__ __forceinline__ void wave_argmax32(float m, int pk, float& mo, int& po) {
#pragma unroll
  for (int o = 16; o > 0; o >>= 1) {
    float om = __shfl_xor(m, o, 32);
    int   op = __shfl_xor(pk, o, 32);
    if (om > m || (om == m && op > pk)) { m = om; pk = op; }
  }
  mo = m; po = pk;
}

// atan2(|v1 x v2|, v1.v2), negated where dot < 0
__device__ __forceinline__ float signed_angle(float ax, float ay, float az,
                                              float bx, float by, float bz) {
  float cx = ay * bz - az * by;
  float cy = az * bx - ax * bz;
  float cz = ax * by - ay * bx;
  float s  = sqrtf(cx * cx + cy * cy + cz * cz);
  float c  = ax * bx + ay * by + az * bz;
  float a  = atan2f(s, c);
  return (c < 0.0f) ? -a : a;
}

// ---------------- kernel 1: KNN + PPF + WMMA GEMM + max ----------------
__global__ __launch_bounds__(256) void ppf_glob_kernel(
    const float* __restrict__ points, const float* __restrict__ nodes,
    const float* __restrict__ pnrm,   const float* __restrict__ nnrm,
    const float* __restrict__ W,      const float* __restrict__ bias,
    float* __restrict__ out_glob) {
  __shared__ float lds_pts[CHUNK * 3];          // 24 KB point tile
  __shared__ int   lds_knn[WAVES][SLOTS];       // 2 KB compacted neighbor ids

  const int   tid  = threadIdx.x;
  const int   lane = tid & 31;
  const int   w    = tid >> 5;
  const int   g    = blockIdx.x * WAVES + w;    // flattened (b, m)
  const int   b    = g / M_;
  const float INF  = __builtin_inff();

  const float* pb  = points + (size_t)b * N_ * 3;
  const float* pnb = pnrm   + (size_t)b * N_ * 3;

  const float nx  = nodes[g * 3 + 0], ny  = nodes[g * 3 + 1], nz  = nodes[g * 3 + 2];
  const float n1x = nnrm[g * 3 + 0],  n1y = nnrm[g * 3 + 1],  n1z = nnrm[g * 3 + 2];

  // Preload W operand tiles + bias into registers (loop-invariant, 4 KB total).
  // B-tile layout for V_WMMA_F32_16X16X4_F32 (4x16, K x N):
  //   VGPR0: K=0 (lanes 0-15) / K=1 (lanes 16-31); VGPR1: K=2 / K=3; N = lane%16.
  const bool hi      = lane >= 16;
  const int  colbase = lane & 15;
  const int  wr0 = (hi ? 1 : 0) * D_;
  const int  wr1 = (hi ? 3 : 2) * D_;
  v2f   Bt[16];
  float bb[16];
#pragma unroll
  for (int c = 0; c < 16; ++c) {
    const int col = c * 16 + colbase;
    Bt[c].x = W[wr0 + col];
    Bt[c].y = W[wr1 + col];
    bb[c]   = bias[col];
  }

  // streaming top-64 (2 slots per lane)
  float d0 = INF, d1 = INF;
  int   i0 = 0,   i1 = 0;
  float thr = INF;

  for (int start = 0; start < N_; start += CHUNK) {
    const int cn = min(CHUNK, N_ - start);
    const int nf = cn * 3;
#if USE_ASYNC_LDS
    for (int j = tid; j < nf; j += 256) {
      __builtin_amdgcn_global_load_async_to_lds_b32(
          (gint_ptr)(pb + (size_t)start * 3 + j),
          (lint_ptr)(&lds_pts[j]), 0, 0);
    }
    __builtin_amdgcn_s_wait_asynccnt(0);
#else
    for (int j = tid; j < nf; j += 256) lds_pts[j] = pb[(size_t)start * 3 + j];
#endif
    __syncthreads();

    for (int base2 = 0; base2 < cn; base2 += 32) {
      const int i = base2 + lane;
      float d2c = INF;
      if (i < cn) {
        float px = lds_pts[3 * i], py = lds_pts[3 * i + 1], pz = lds_pts[3 * i + 2];
        float dx = nx - px, dy = ny - py, dz = nz - pz;
        d2c = dx * dx + dy * dy + dz * dz;
      }
      unsigned mask = (unsigned)__ballot(d2c < thr);
      while (mask) {
        const int src = __ffs(mask) - 1;
        mask &= mask - 1;
        const float dc = __shfl(d2c, src, 32);
        const int   ic = start + base2 + src;
        if (dc < thr) {                       // thr may have dropped
          float mm = fmaxf(d0, d1);
          int   pk = (lane << 1) | (d1 > d0 ? 1 : 0);
          float bm; int bp;
          wave_argmax32(mm, pk, bm, bp);
          if ((bp >> 1) == lane) {
            if (bp & 1) { d1 = dc; i1 = ic; } else { d0 = dc; i0 = ic; }
          }
          thr = wave_max32(fmaxf(d0, d1));
        }
      }
    }
    __syncthreads();
  }

  // drop the 14 largest -> exact top-50 (as a set; order irrelevant under max)
  bool v0 = true, v1 = true;
#pragma unroll 1
  for (int r = 0; r < DROP; ++r) {
    float m0 = v0 ? d0 : -INF, m1 = v1 ? d1 : -INF;
    float mm = fmaxf(m0, m1);
    int   pk = (lane << 1) | (m1 > m0 ? 1 : 0);
    float bm; int bp;
    wave_argmax32(mm, pk, bm, bp);
    if ((bp >> 1) == lane) { if (bp & 1) v1 = false; else v0 = false; }
  }

  // ballot compaction into LDS: 50 valid ids, pad to 64 with entry 0
  const unsigned msk0  = (unsigned)__ballot(v0);
  const unsigned msk1  = (unsigned)__ballot(v1);
  const unsigned below = (1u << lane) - 1u;
  const int base1 = __popc(msk0);
  if (v0) lds_knn[w][__popc(msk0 & below)]         = i0;
  if (v1) lds_knn[w][base1 + __popc(msk1 & below)] = i1;
  __syncthreads();
  const int count  = base1 + __popc(msk1);          // == 50
  const int ifirst = lds_knn[w][0];
  for (int j = count + lane; j < SLOTS; j += 32) lds_knn[w][j] = ifirst;
  __syncthreads();

  // PPF features -> WMMA A tiles (16x4 f32 layout: lanes<16 hold feats 0,1; lanes>=16 feats 2,3)
  v2f At[4];
#pragma unroll
  for (int t = 0; t < 4; ++t) {
    const int row = t * 16 + colbase;
    const int idx = lds_knn[w][row];
    const float px  = pb[idx * 3 + 0],  py  = pb[idx * 3 + 1],  pz  = pb[idx * 3 + 2];
    const float qx  = pnb[idx * 3 + 0], qy  = pnb[idx * 3 + 1], qz  = pnb[idx * 3 + 2];
    const float ix = px - nx, iy = py - ny, iz = pz - nz;
    const float d   = sqrtf(ix * ix + iy * iy + iz * iz);          // SCALE == 1
    const float f1  = signed_angle(ix, iy, iz, n1x, n1y, n1z);     // angle(ij, n1)
    const float f2  = signed_angle(-ix, -iy, -iz, qx, qy, qz);     // angle(-ij, n2)
    const float f3  = signed_angle(n1x, n1y, n1z, qx, qy, qz);     // angle(n1, n2)
    At[t].x = hi ? f2 : d;
    At[t].y = hi ? f3 : f1;
  }

  // D = A(16x4) x W(4x16) per 16-col chunk; max over 64 rows; +bias at end.
  // Fully unrolled: 64 independent WMMAs, no memory ops between them.
#pragma unroll
  for (int c = 0; c < 16; ++c) {
    const int col = c * 16 + colbase;
    v8f z = {};
    v8f r0 = __builtin_amdgcn_wmma_f32_16x16x4_f32(false, At[0], false, Bt[c], (short)0, z, false, false);
    v8f r1 = __builtin_amdgcn_wmma_f32_16x16x4_f32(false, At[1], false, Bt[c], (short)0, z, false, false);
    v8f r2 = __builtin_amdgcn_wmma_f32_16x16x4_f32(false, At[2], false, Bt[c], (short)0, z, false, false);
    v8f r3 = __builtin_amdgcn_wmma_f32_16x16x4_f32(false, At[3], false, Bt[c], (short)0, z, false, false);
    float cm = -INF;
#pragma unroll
    for (int k = 0; k < 8; ++k)
      cm = fmaxf(cm, fmaxf(fmaxf(r0[k], r1[k]), fmaxf(r2[k], r3[k])));
    cm = fmaxf(cm, __shfl_xor(cm, 16, 32));       // fold lane halves (rows m and m+8)
    if (lane < 16) out_glob[(size_t)g * D_ + col] = cm + bb[c];
  }
}

// ---------------- kernel 2: sinusoidal position embedding ----------------
__global__ __launch_bounds__(256) void pos_emb_kernel(const float* __restrict__ nodes,
                                                      float* __restrict__ out_pos) {
  const int bm = blockIdx.x;       // 0 .. B*M-1
  const int d  = threadIdx.x;      // 0 .. 255
  float v = 0.0f;
  if (d < 3 * NPF) {
    const int c = d / NPF;
    const int r = d - c * NPF;
    const int i = r >> 1;
    const float x = nodes[bm * 3 + c];
    const float p = x * __powf(10000.0f, -(2.0f * (float)i) / (float)NPF);
    v = (r & 1) ? __cosf(p) : __sinf(p);
  }
  out_pos[(size_t)bm * D_ + d] = v;
}

extern "C" void kernel_launch(void* const* d_in, const int* in_sizes, int n_in,
                              void* d_out, int out_size, void* d_ws, size_t ws_size,
                              hipStream_t stream) {
  (void)in_sizes; (void)n_in; (void)out_size; (void)d_ws; (void)ws_size;
  const float* points = (const float*)d_in[0];
  const float* nodes  = (const float*)d_in[1];
  const float* pnrm   = (const float*)d_in[2];
  const float* nnrm   = (const float*)d_in[3];
  const float* W      = (const float*)d_in[4];
  const float* bias   = (const float*)d_in[5];
  float* out_glob = (float*)d_out;
  float* out_pos  = out_glob + (size_t)B_ * M_ * D_;

  ppf_glob_kernel<<<(B_ * M_) / WAVES, 256, 0, stream>>>(points, nodes, pnrm, nnrm, W, bias, out_glob);
  pos_emb_kernel<<<B_ * M_, 256, 0, stream>>>(nodes, out_pos);
}